// Weatherformer_24687472018044
// MI455X (gfx1250) — compile-verified
//
#include <hip/hip_runtime.h>
#include <hip/hip_bf16.h>

typedef __attribute__((ext_vector_type(2))) float v2f;
typedef __attribute__((ext_vector_type(8))) float v8f;

#define WMMA4(a, b, c) \
    __builtin_amdgcn_wmma_f32_16x16x4_f32(false, (a), false, (b), (short)0, (c), false, false)

// ---------------------------------------------------------------------------
// Model constants
// ---------------------------------------------------------------------------
#define BB 8
#define SS 1024
#define EE 64
#define HH 8
#define HD 8
#define FF 256
#define INF_ 31
#define NTOK (BB * SS)         // 8192
#define NBH (BB * HH)          // 64
#define INV_SQRT_HD 0.35355339059327373f

// ---------------------------------------------------------------------------
// Embedding + positional encoding: x[t,e] = weather[t,:]@W_emb + b_emb + pos
// grid 8192 x block 64 (tiny fraction of total work -> VALU is fine)
// ---------------------------------------------------------------------------
__global__ void embed_pos_kernel(const float* __restrict__ weather,
                                 const float* __restrict__ coords,
                                 const float* __restrict__ W_emb,
                                 const float* __restrict__ b_emb,
                                 float* __restrict__ x) {
    int t = blockIdx.x;            // token 0..8191
    int e = threadIdx.x;           // channel 0..63
    int b = t >> 10, s = t & 1023;
    const float* wrow = weather + t * INF_;
    float acc = b_emb[e];
#pragma unroll
    for (int i = 0; i < INF_; ++i) acc += wrow[i] * W_emb[i * EE + e];
    int g = e & 3, j = e >> 2;
    float div = __expf((float)(4 * j) * (-9.210340371976184f / 64.0f)); // -ln(1e4)/E
    float lat = coords[b * 2 + 0] * (3.14159265358979323846f / 180.0f);
    float lon = coords[b * 2 + 1] * (3.14159265358979323846f / 180.0f);
    float p;
    if (g == 0)      p = __sinf((float)s * div);
    else if (g == 1) p = __cosf((float)s * div);
    else if (g == 2) p = __sinf(lat * div);
    else             p = __cosf(lon * div);
    x[t * EE + e] = acc + p;
}

// ---------------------------------------------------------------------------
// Generic fp32 WMMA GEMM: C[M,N] = A[M,K] @ B[K,N] + bias (+optional ReLU)
// One wave (32 lanes) per 16x16 output tile, K-steps of 4 via
// v_wmma_f32_16x16x4_f32. Fragment layouts per CDNA5 ISA 7.12.2.
// ---------------------------------------------------------------------------
__device__ __forceinline__ v8f wmma_tile_f32(const float* __restrict__ A,
                                             const float* __restrict__ B,
                                             int tm, int tn, int N, int K,
                                             int lane) {
    int ln = lane & 15;
    int hi = lane >> 4;                       // half-wave: K pair {0,1} vs {2,3}
    v8f acc = {};
    const float* arow = A + (tm + ln) * K + hi * 2;
    const float* bcol = B + tn + ln + hi * 2 * N;
    for (int k0 = 0; k0 < K; k0 += 4) {
        v2f a, b;
        a.x = arow[k0];
        a.y = arow[k0 + 1];
        b.x = bcol[k0 * N];
        b.y = bcol[(k0 + 1) * N];
        acc = WMMA4(a, b, acc);
    }
    return acc;
}

__global__ void gemm_bias_kernel(const float* __restrict__ A,
                                 const float* __restrict__ B,
                                 const float* __restrict__ bias,
                                 float* __restrict__ C,
                                 int M, int N, int K, int relu) {
    int tn = blockIdx.x * 16;
    int tm = blockIdx.y * 16;
    int lane = threadIdx.x;
    int ln = lane & 15, hi = lane >> 4;
    v8f acc = wmma_tile_f32(A, B, tm, tn, N, K, lane);
    int col = tn + ln;
    float bs = bias[col];
#pragma unroll
    for (int r = 0; r < 8; ++r) {
        int row = tm + r + hi * 8;
        float v = acc[r] + bs;
        if (relu) v = fmaxf(v, 0.0f);
        C[row * N + col] = v;
    }
}

// ---------------------------------------------------------------------------
// QKV GEMM (M=8192, N=192, K=64) with scatter epilogue implementing torch's
// [3,B,S,E] -> [3,S,B*H,HD] reinterpretation; Q/K get granularity embedding,
// Q additionally pre-scaled by 1/sqrt(HD). Output: [bh][s'][hd] contiguous.
// ---------------------------------------------------------------------------
__global__ void qkv_kernel(const float* __restrict__ x,
                           const float* __restrict__ Wqkv,
                           const float* __restrict__ bqkv,
                           const float* __restrict__ gran_l,   // [NG,E]
                           const int* __restrict__ temporal,   // [B,2]
                           float* __restrict__ Q,
                           float* __restrict__ K,
                           float* __restrict__ V) {
    int tn = blockIdx.x * 16;
    int tm = blockIdx.y * 16;
    int lane = threadIdx.x;
    int ln = lane & 15, hi = lane >> 4;
    v8f acc = wmma_tile_f32(x, Wqkv, tm, tn, 3 * EE, EE, lane);
    int col = tn + ln;               // 0..191
    int which = col >> 6;            // uniform per block (16 | 64)
    int e = col & 63;
    float bs = bqkv[col];
#pragma unroll
    for (int r = 0; r < 8; ++r) {
        int row = tm + r + hi * 8;           // token
        int b = row >> 10, s = row & 1023;
        int flat = (b << 16) + (s << 6) + e; // index into (B,S,E) flattening
        int s2 = flat >> 9;                  // reinterpreted seq index
        int bh = (flat >> 3) & 63;           // reinterpreted batch-head
        int hd = flat & 7;
        int dst = bh * (SS * HD) + s2 * HD + hd;
        float v = acc[r] + bs;
        if (which == 0) {
            int b2 = bh >> 3;
            float ge = gran_l[temporal[b2 * 2 + 1] * EE + ((bh & 7) * 8 + hd)];
            Q[dst] = (v + ge) * INV_SQRT_HD;
        } else if (which == 1) {
            int b2 = bh >> 3;
            float ge = gran_l[temporal[b2 * 2 + 1] * EE + ((bh & 7) * 8 + hd)];
            K[dst] = v + ge;
        } else {
            V[dst] = v;
        }
    }
}

// ---------------------------------------------------------------------------
// Flash attention, one wave per (bh, 16-query tile). Computes S^T = K @ Q^T
// so each lane owns ONE query column: online-softmax state (m,l) is a scalar
// per lane; k-reduction is 8 register ops + one shfl_xor(16). P^T tile feeds
// O^T = V^T @ P^T with 8 cross-half shuffles/tile (no LDS transpose).
// Result scattered directly into torch's concat layout Oc[8192][64].
// ---------------------------------------------------------------------------
__global__ void attn_kernel(const float* __restrict__ Q,
                            const float* __restrict__ K,
                            const float* __restrict__ V,
                            float* __restrict__ Oc) {
    int bh = blockIdx.y;
    int q0 = blockIdx.x * 16;
    int lane = threadIdx.x;
    int ln = lane & 15, hi = lane >> 4;
    const float* Qb = Q + bh * (SS * HD);
    const float* Kb = K + bh * (SS * HD);
    const float* Vb = V + bh * (SS * HD);

    // B-fragments of Q^T (persist across all k-tiles)
    const float* qp = Qb + (q0 + ln) * HD + hi * 2;
    v2f bq0, bq1;
    bq0.x = qp[0]; bq0.y = qp[1];
    bq1.x = qp[4]; bq1.y = qp[5];

    float m = -1e30f, l = 0.0f;
    v8f o = {};
    float vmsk = (ln < 8) ? 1.0f : 0.0f;     // zero-pad V^T rows 8..15

    for (int kt = 0; kt < SS; kt += 16) {
        // A-fragments of K tile
        const float* kp = Kb + (kt + ln) * HD + hi * 2;
        v2f ak0, ak1;
        ak0.x = kp[0]; ak0.y = kp[1];
        ak1.x = kp[4]; ak1.y = kp[5];

        v8f sc = {};
        sc = WMMA4(ak0, bq0, sc);            // S^T[k][q], rows=k, cols=q
        sc = WMMA4(ak1, bq1, sc);

        // online softmax over k for this lane's query column
        float tmax = sc[0];
#pragma unroll
        for (int r = 1; r < 8; ++r) tmax = fmaxf(tmax, sc[r]);
        tmax = fmaxf(tmax, __shfl_xor(tmax, 16, 32));
        float mnew = fmaxf(m, tmax);
        float corr = __expf(m - mnew);
        float pv[8];
        float tl = 0.0f;
#pragma unroll
        for (int r = 0; r < 8; ++r) {
            float ev = __expf(sc[r] - mnew);
            pv[r] = ev;
            tl += ev;
        }
        tl += __shfl_xor(tl, 16, 32);
        l = l * corr + tl;
        m = mnew;
#pragma unroll
        for (int r = 0; r < 8; ++r) o[r] *= corr;

        // O^T += V^T @ P^T : four 16x16x4 steps over this 16-wide k tile
#pragma unroll
        for (int t = 0; t < 4; ++t) {
            int kk = kt + 4 * t + 2 * hi;
            v2f av;
            av.x = Vb[kk * HD + (ln & 7)] * vmsk;
            av.y = Vb[(kk + 1) * HD + (ln & 7)] * vmsk;
            // B-frag of P^T rows 4t..4t+3 for col ln: mix own + partner half
            const int base = (t & 1) * 4;
            const int owner = t >> 1;
            float send0 = hi ? pv[base]     : pv[base + 2];
            float send1 = hi ? pv[base + 1] : pv[base + 3];
            float rec0 = __shfl_xor(send0, 16, 32);
            float rec1 = __shfl_xor(send1, 16, 32);
            float own0 = hi ? pv[base + 2] : pv[base];
            float own1 = hi ? pv[base + 3] : pv[base + 1];
            v2f bp;
            bool ot = (hi == owner);
            bp.x = ot ? own0 : rec0;
            bp.y = ot ? own1 : rec1;
            o = WMMA4(av, bp, o);            // rows=hd, cols=q
        }
    }

    // low half lanes hold hd=0..7 for query column ln; normalize + scatter
    if (hi == 0) {
        float inv = 1.0f / l;
        int s2 = q0 + ln;
        int row = s2 * HH + (bh >> 3);       // concat-layout row in [8192]
        float* dst = Oc + row * EE + (bh & 7) * HD;
#pragma unroll
        for (int r = 0; r < 8; ++r) dst[r] = o[r] * inv;
    }
}

// ---------------------------------------------------------------------------
// Fused residual + LayerNorm over E=64, one wave per row, butterfly reduce.
// x <- LN(x + a) * g + be
// ---------------------------------------------------------------------------
__global__ void add_ln_kernel(float* __restrict__ x,
                              const float* __restrict__ a,
                              const float* __restrict__ g,
                              const float* __restrict__ be) {
    int row = blockIdx.x * (blockDim.x >> 5) + (threadIdx.x >> 5);
    int lane = threadIdx.x & 31;
    const float2 xv = *(const float2*)(x + row * EE + lane * 2);
    const float2 av = *(const float2*)(a + row * EE + lane * 2);
    float u0 = xv.x + av.x, u1 = xv.y + av.y;
    float s = u0 + u1;
    float ss = u0 * u0 + u1 * u1;
#pragma unroll
    for (int msk = 1; msk < 32; msk <<= 1) {
        s += __shfl_xor(s, msk, 32);
        ss += __shfl_xor(ss, msk, 32);
    }
    float mean = s * (1.0f / 64.0f);
    float var = ss * (1.0f / 64.0f) - mean * mean;
    float inv = rsqrtf(var + 1e-5f);
    float2 out;
    out.x = (u0 - mean) * inv * g[lane * 2] + be[lane * 2];
    out.y = (u1 - mean) * inv * g[lane * 2 + 1] + be[lane * 2 + 1];
    *(float2*)(x + row * EE + lane * 2) = out;
}

// ---------------------------------------------------------------------------
// Final projection E=64 -> OUT=31 (tiny; VALU)
// ---------------------------------------------------------------------------
__global__ void fc_kernel(const float* __restrict__ x,
                          const float* __restrict__ W,
                          const float* __restrict__ b,
                          float* __restrict__ out) {
    int row = blockIdx.x;
    int c = threadIdx.x;
    if (c < 31) {
        float acc = b[c];
        const float* xr = x + row * EE;
#pragma unroll
        for (int i = 0; i < EE; ++i) acc += xr[i] * W[i * 31 + c];
        out[row * 31 + c] = acc;
    }
}

// ---------------------------------------------------------------------------
// Launch
// ---------------------------------------------------------------------------
extern "C" void kernel_launch(void* const* d_in, const int* in_sizes, int n_in,
                              void* d_out, int out_size, void* d_ws, size_t ws_size,
                              hipStream_t stream) {
    const float* weather  = (const float*)d_in[0];
    const float* coords   = (const float*)d_in[1];
    const int*   temporal = (const int*)d_in[2];
    const float* W_emb = (const float*)d_in[3];
    const float* b_emb = (const float*)d_in[4];
    const float* W_qkv = (const float*)d_in[5];
    const float* b_qkv = (const float*)d_in[6];
    const float* W_out = (const float*)d_in[7];
    const float* b_out = (const float*)d_in[8];
    const float* gran  = (const float*)d_in[9];
    const float* g1  = (const float*)d_in[10];
    const float* be1 = (const float*)d_in[11];
    const float* W1  = (const float*)d_in[12];
    const float* b1  = (const float*)d_in[13];
    const float* W2  = (const float*)d_in[14];
    const float* b2  = (const float*)d_in[15];
    const float* g2  = (const float*)d_in[16];
    const float* be2 = (const float*)d_in[17];
    const float* W_fc = (const float*)d_in[18];
    const float* b_fc = (const float*)d_in[19];
    float* out = (float*)d_out;

    // workspace layout (floats)
    float* ws  = (float*)d_ws;
    float* x   = ws;                        // 8192*64
    float* Qb  = x   + NTOK * EE;           // 64*1024*8
    float* Kb  = Qb  + NBH * SS * HD;
    float* Vb  = Kb  + NBH * SS * HD;
    float* Oc  = Vb  + NBH * SS * HD;       // 8192*64
    float* tmp = Oc  + NTOK * EE;           // 8192*64
    float* hid = tmp + NTOK * EE;           // 8192*256

    embed_pos_kernel<<<NTOK, EE, 0, stream>>>(weather, coords, W_emb, b_emb, x);

    for (int l = 0; l < 3; ++l) {
        qkv_kernel<<<dim3((3 * EE) / 16, NTOK / 16), 32, 0, stream>>>(
            x, W_qkv + l * EE * 3 * EE, b_qkv + l * 3 * EE,
            gran + l * 31 * EE, temporal, Qb, Kb, Vb);

        attn_kernel<<<dim3(SS / 16, NBH), 32, 0, stream>>>(Qb, Kb, Vb, Oc);

        gemm_bias_kernel<<<dim3(EE / 16, NTOK / 16), 32, 0, stream>>>(
            Oc, W_out + l * EE * EE, b_out + l * EE, tmp, NTOK, EE, EE, 0);

        add_ln_kernel<<<NTOK / 8, 256, 0, stream>>>(x, tmp, g1 + l * EE, be1 + l * EE);

        gemm_bias_kernel<<<dim3(FF / 16, NTOK / 16), 32, 0, stream>>>(
            x, W1 + l * EE * FF, b1 + l * FF, hid, NTOK, FF, EE, 1);

        gemm_bias_kernel<<<dim3(EE / 16, NTOK / 16), 32, 0, stream>>>(
            hid, W2 + l * FF * EE, b2 + l * EE, tmp, NTOK, EE, FF, 0);

        add_ln_kernel<<<NTOK / 8, 256, 0, stream>>>(x, tmp, g2 + l * EE, be2 + l * EE);
    }

    fc_kernel<<<NTOK, 32, 0, stream>>>(x, W_fc, b_fc, out);
}